// IrrepsLinear_19078244729272
// MI455X (gfx1250) — compile-verified
//
#include <hip/hip_runtime.h>

typedef __attribute__((ext_vector_type(2))) float v2f;
typedef __attribute__((ext_vector_type(8))) float v8f;

#define N_SAMPLES 50000
#define FEAT      1152
#define MUL       128                  // multiplicity (K and #cols per segment)
#define NB        16                   // samples per block-iteration
#define NBLKS     (N_SAMPLES / NB)     // 3125 (exact)
#define GRIDX     625                  // each WG loops over 5 sample-blocks
#define STRIDE_M  (MUL + 4)            // 132 floats; 132 % 64 == 4 -> conflict-free b64 reads

// LDS layout after staging: lds[(nl*D + j) * STRIDE_M + m] = x[n0+nl][OFF + m*D + j]
// i.e. each GEMM row r = nl*D + j is contiguous over m -> A K-pairs are ds_load_b64
// with immediate offsets.

template<int D, int OFF>
__device__ __forceinline__
void seg_body(const float* __restrict__ x, const float* __restrict__ W,
              float* __restrict__ out, float* lds)
{
    const int tid   = threadIdx.x;          // 0..255
    const int wave  = tid >> 5;             // 0..7 (wave32)
    const int lane  = tid & 31;
    const int l16   = lane & 15;
    const int khalf = (lane >> 4) << 1;     // K pair base: 0 (lanes 0-15) / 2 (lanes 16-31)

    // ---- B fragments: this wave's 16 columns of W, VGPR-resident across all blocks ----
    // B 4x16 layout: VGPR0 = row K=khalf, VGPR1 = row K=khalf+1, N = l16.
    float bfrag[32][2];
    {
        const int col = wave * 16 + l16;
        #pragma unroll
        for (int s = 0; s < 32; ++s) {
            const int m0 = 4 * s + khalf;
            bfrag[s][0] = W[(size_t)m0 * MUL + col];
            bfrag[s][1] = W[(size_t)(m0 + 1) * MUL + col];
        }
    }

    const float alpha = 0.08838834764831845f;   // 1/sqrt(128)

    for (int blk = blockIdx.x; blk < NBLKS; blk += GRIDX) {
        const int n0 = blk * NB;

        __syncthreads();   // LDS reuse guard between block-iterations

        // ---- Stage 1: coalesced b128 global reads, transposed LDS writes ----
        #pragma unroll
        for (int it = 0; it < 2 * D; ++it) {
            const int e4 = it * 256 + tid;
            const int e  = e4 << 2;
            const int nl = e / (MUL * D);
            const int el = e - nl * (MUL * D);
            const float4 v = *(const float4*)(x + (size_t)(n0 + nl) * FEAT + OFF + el);
            #pragma unroll
            for (int q = 0; q < 4; ++q) {
                const int ee = el + q;
                const int m  = ee / D;           // compile-time D -> shifts/mulhi
                const int j  = ee - m * D;
                lds[(nl * D + j) * STRIDE_M + m] = ((const float*)&v)[q];
            }
        }

        __syncthreads();

        // ---- Stage 2: D row-tiles of 16 rows; 32 chained WMMAs over K=128 ----
        #pragma unroll
        for (int t = 0; t < D; ++t) {
            // A 16x4 layout: lane holds row M=l16, Ks {khalf, khalf+1} -> contiguous b64
            const float* __restrict__ aBase = &lds[(t * 16 + l16) * STRIDE_M + khalf];

            v8f acc = {};
            #pragma unroll
            for (int s = 0; s < 32; ++s) {
                const v2f a = *(const v2f*)(aBase + 4 * s);   // ds_load_b64 offset:16*s
                v2f b;
                b.x = bfrag[s][0];
                b.y = bfrag[s][1];
                acc = __builtin_amdgcn_wmma_f32_16x16x4_f32(
                    /*neg_a=*/false, a, /*neg_b=*/false, b,
                    /*c_mod=*/(short)0, acc, /*reuse_a=*/false, /*reuse_b=*/false);
            }

            // C/D layout: lane holds col N=l16; VGPR v = row (lane<16 ? v : v+8)
            const int o = wave * 16 + l16;
            #pragma unroll
            for (int v = 0; v < 8; ++v) {
                const int M   = ((lane >> 4) << 3) + v;
                const int rr  = t * 16 + M;
                const int nl2 = rr / D;
                const int jj  = rr - nl2 * D;
                out[(size_t)(n0 + nl2) * FEAT + OFF + o * D + jj] = acc[v] * alpha;
            }
        }
    }
}

__global__ __launch_bounds__(256, 1)
void irreps_linear_wmma(const float* __restrict__ x,
                        const float* __restrict__ W0,
                        const float* __restrict__ W1,
                        const float* __restrict__ W2,
                        float* __restrict__ out)
{
    __shared__ float lds[NB * 5 * STRIDE_M];   // 80 rows * 132 floats = 42240 B

    const int seg = blockIdx.y;
    if (seg == 0) {
        seg_body<1, 0  >(x, W0, out, lds);
    } else if (seg == 1) {
        seg_body<3, 128>(x, W1, out, lds);
    } else {
        seg_body<5, 512>(x, W2, out, lds);
    }
}

extern "C" void kernel_launch(void* const* d_in, const int* in_sizes, int n_in,
                              void* d_out, int out_size, void* d_ws, size_t ws_size,
                              hipStream_t stream) {
    const float* x  = (const float*)d_in[0];
    const float* W0 = (const float*)d_in[1];
    const float* W1 = (const float*)d_in[2];
    const float* W2 = (const float*)d_in[3];
    float* out = (float*)d_out;

    dim3 grid(GRIDX, 3);    // 625 WGs x 3 segments; each WG covers 5 sample-blocks
    dim3 block(256);        // 8 wave32 waves
    irreps_linear_wmma<<<grid, block, 0, stream>>>(x, W0, W1, W2, out);
}